// LuongAttention_50663434223730
// MI455X (gfx1250) — compile-verified
//
#include <hip/hip_runtime.h>
#include <hip/hip_bf16.h>
#include <math.h>

// LuongAttention, H=1024, S=32768, fp32.
// Memory-bound (~0.5 FLOP/byte): two streaming passes over enc (128MB each),
// scores pass via V_WMMA_F32_16X16X4_F32, context pass via float4 VALU FMA.

#define H 1024
#define S 32768
#define ROWS_PER_BLOCK 128   // 8 waves * 16 rows (scores pass)
#define CHUNK 128            // rows per block (context pass)
#define NBLK_S (S / ROWS_PER_BLOCK)   // 256
#define NBLK_C (S / CHUNK)            // 256

typedef __attribute__((ext_vector_type(2))) float v2f;
typedef __attribute__((ext_vector_type(4))) float v4f;
typedef __attribute__((ext_vector_type(8))) float v8f;

// ---------------------------------------------------------------- q = W^T h
__global__ void k_q(const float* __restrict__ h, const float* __restrict__ W,
                    float* __restrict__ q) {
    __shared__ float h_sh[H];
    const int tid = threadIdx.x;
    for (int i = tid; i < H; i += 256) h_sh[i] = h[i];
    __syncthreads();
    const int j = blockIdx.x * 256 + tid;          // column, coalesced
    float acc = 0.0f;
    for (int i = 0; i < H; ++i)
        acc = fmaf(h_sh[i], W[(size_t)i * H + j], acc);
    q[j] = acc;
}

// ----------------------------------------------- scores = enc @ q via WMMA
// A (16x4 f32, ISA layout): lane L holds row M=L&15; VGPR0/1 hold K = koff,
// koff+1 with koff = 2*(L>>4).  B (4x16): all 16 columns equal q[k], mirrored
// K split across half-waves.  D: lane 0 holds M=0..7 in c[0..7] (N=0),
// lane 16 holds M=8..15.
__global__ void k_scores(const float* __restrict__ enc, const float* __restrict__ q,
                         float* __restrict__ scores, float* __restrict__ blockmax) {
    __shared__ float q_sh[H];
    __shared__ float wmax[16];
    const int tid = threadIdx.x;
    for (int i = tid; i < H; i += 256) q_sh[i] = q[i];
    __syncthreads();

    const int lane    = tid & 31;
    const int wave    = tid >> 5;
    const int rowBase = blockIdx.x * ROWS_PER_BLOCK + wave * 16;
    const int m       = lane & 15;
    const int koff    = (lane >> 4) << 1;          // 0 or 2
    const float* rowPtr = enc + (size_t)(rowBase + m) * H + koff;

    v8f c = {};
    for (int kb = 0; kb < H; kb += 4) {
        v2f a = *(const v2f*)(rowPtr + kb);            // global_load_b64
        v2f b = *(const v2f*)(&q_sh[kb + koff]);       // ds_load_b64
        c = __builtin_amdgcn_wmma_f32_16x16x4_f32(
                /*neg_a=*/false, a, /*neg_b=*/false, b,
                /*c_mod=*/(short)0, c, /*reuse_a=*/false, /*reuse_b=*/false);
    }

    if (m == 0) {                                   // lanes 0 and 16 hold D col N=0
        const int half = lane >> 4;
        float mx = c[0];
        #pragma unroll
        for (int r = 0; r < 8; ++r) {
            scores[rowBase + half * 8 + r] = c[r];
            mx = fmaxf(mx, c[r]);
        }
        wmax[wave * 2 + half] = mx;
    }
    __syncthreads();
    if (tid == 0) {
        float mx = wmax[0];
        for (int i = 1; i < 16; ++i) mx = fmaxf(mx, wmax[i]);
        blockmax[blockIdx.x] = mx;
    }
}

// ------------------------------------------------------- global max reduce
__global__ void k_maxred(const float* __restrict__ blockmax, float* __restrict__ gmax) {
    __shared__ float sh[256];
    const int tid = threadIdx.x;
    sh[tid] = blockmax[tid];
    __syncthreads();
    for (int off = 128; off > 0; off >>= 1) {
        if (tid < off) sh[tid] = fmaxf(sh[tid], sh[tid + off]);
        __syncthreads();
    }
    if (tid == 0) gmax[0] = sh[0];
}

// --------------- e = exp(score - max); block sums; partial context (pass 2)
__global__ void k_exp_pctx(const float* __restrict__ enc, const float* __restrict__ scores,
                           const float* __restrict__ gmax, float* __restrict__ expv,
                           float* __restrict__ blocksum, float* __restrict__ pctx) {
    __shared__ float e_sh[CHUNK];
    const int tid = threadIdx.x;
    const int s0  = blockIdx.x * CHUNK;
    const float gm = gmax[0];
    if (tid < CHUNK) {
        const float e = expf(scores[s0 + tid] - gm);
        e_sh[tid] = e;
        expv[s0 + tid] = e;
    }
    __syncthreads();
    if (tid == 0) {                                 // fixed-order (deterministic)
        float sum = 0.0f;
        for (int i = 0; i < CHUNK; ++i) sum += e_sh[i];
        blocksum[blockIdx.x] = sum;
    }
    // each thread owns 4 contiguous columns -> global_load_b128, fully coalesced
    v4f acc = {0.0f, 0.0f, 0.0f, 0.0f};
    const v4f* encv = (const v4f*)(enc + (size_t)s0 * H);
    for (int s = 0; s < CHUNK; ++s) {
        const float e = e_sh[s];
        const v4f  x = encv[s * (H / 4) + tid];
        acc += x * e;
    }
    *(v4f*)(pctx + (size_t)blockIdx.x * H + tid * 4) = acc;
}

// ------------------------------------------------------------- 1/sum reduce
__global__ void k_sumred(const float* __restrict__ blocksum, float* __restrict__ inv) {
    __shared__ float sh[256];
    const int tid = threadIdx.x;
    sh[tid] = blocksum[tid];
    __syncthreads();
    for (int off = 128; off > 0; off >>= 1) {       // fixed-order tree
        if (tid < off) sh[tid] += sh[tid + off];
        __syncthreads();
    }
    if (tid == 0) inv[0] = 1.0f / sh[0];
}

// ----------------------------------------------- context = (sum pctx) * inv
__global__ void k_ctx(const float* __restrict__ pctx, const float* __restrict__ inv,
                      float* __restrict__ out) {
    const int j = blockIdx.x * 256 + threadIdx.x;
    float s = 0.0f;
    for (int b = 0; b < NBLK_C; ++b) s += pctx[(size_t)b * H + j];
    out[j] = s * inv[0];
}

// ----------------------------------------------------- attn = expv * inv
__global__ void k_attn(const float* __restrict__ expv, const float* __restrict__ inv,
                       float* __restrict__ out) {
    const int s = blockIdx.x * 256 + threadIdx.x;
    out[H + s] = expv[s] * inv[0];
}

extern "C" void kernel_launch(void* const* d_in, const int* in_sizes, int n_in,
                              void* d_out, int out_size, void* d_ws, size_t ws_size,
                              hipStream_t stream) {
    const float* h   = (const float*)d_in[0];   // (H,1)
    const float* enc = (const float*)d_in[1];   // (S,H,1)
    const float* W   = (const float*)d_in[2];   // (H,H)
    float* out = (float*)d_out;                 // [context(1024) | attn(32768)]

    // workspace layout (floats)
    float* ws       = (float*)d_ws;
    float* q        = ws;                       // 1024
    float* scores   = ws + 1024;                // 32768
    float* expv     = ws + 34816;               // 32768
    float* blockmax = ws + 67584;               // 256
    float* blocksum = ws + 67840;               // 256
    float* gmax     = ws + 68096;               // 1
    float* inv      = ws + 68097;               // 1
    float* pctx     = ws + 68608;               // 256*1024 = 262144 (16B aligned)

    k_q       <<<H / 256, 256, 0, stream>>>(h, W, q);
    k_scores  <<<NBLK_S,  256, 0, stream>>>(enc, q, scores, blockmax);
    k_maxred  <<<1,       256, 0, stream>>>(blockmax, gmax);
    k_exp_pctx<<<NBLK_C,  256, 0, stream>>>(enc, scores, gmax, expv, blocksum, pctx);
    k_sumred  <<<1,       256, 0, stream>>>(blocksum, inv);
    k_ctx     <<<H / 256, 256, 0, stream>>>(pctx, inv, out);
    k_attn    <<<S / 256, 256, 0, stream>>>(expv, inv, out);
}